// CompressedHE_13297218748877
// MI455X (gfx1250) — compile-verified
//
#include <hip/hip_runtime.h>
#include <hip/hip_bf16.h>

#define HW      (1024 * 1024)
#define NCH     48          // B*C = 16*3
#define NBINS   256
#define BPC     64          // blocks per channel for streaming passes
#define SEG     (HW / BPC)  // 16384 floats per block
#define NVEC    (SEG / 4)   // float4s per block

typedef float v2f __attribute__((ext_vector_type(2)));
typedef float v8f __attribute__((ext_vector_type(8)));

static __device__ __forceinline__ int imin(int a, int b) { return a < b ? a : b; }

// ---------------------------------------------------------------------------
// Kernel 0: zero the global per-channel histograms (d_ws is poisoned 0xAA).
// ---------------------------------------------------------------------------
__global__ void che_zero_hist(int* __restrict__ hist) {
    hist[blockIdx.x * NBINS + threadIdx.x] = 0;
}

// ---------------------------------------------------------------------------
// Kernel 1: per-channel 256-bin histogram.
// 256 threads = 8 waves; one private LDS sub-histogram per wave (8 KB LDS)
// so LDS atomic contention stays within a wave32. Streaming b128 loads.
// ---------------------------------------------------------------------------
__global__ void che_hist(const float* __restrict__ x, int* __restrict__ hist) {
    __shared__ int sub[8][NBINS];
    const int tid = threadIdx.x;
    const int ch  = blockIdx.x / BPC;
    const int blk = blockIdx.x % BPC;

    #pragma unroll
    for (int w = 0; w < 8; ++w) sub[w][tid] = 0;
    __syncthreads();

    int* my = sub[tid >> 5];   // wave-private 256-bin histogram (wave32)
    const float4* p = (const float4*)(x + (size_t)ch * HW + (size_t)blk * SEG);

    for (int i = tid; i < NVEC; i += 256) {
        float4 v = p[i];
        atomicAdd(&my[imin(255, (int)(v.x * 255.0f))], 1);
        atomicAdd(&my[imin(255, (int)(v.y * 255.0f))], 1);
        atomicAdd(&my[imin(255, (int)(v.z * 255.0f))], 1);
        atomicAdd(&my[imin(255, (int)(v.w * 255.0f))], 1);
    }
    __syncthreads();

    int s = 0;
    #pragma unroll
    for (int w = 0; w < 8; ++w) s += sub[w][tid];
    if (s) atomicAdd(&hist[ch * NBINS + tid], s);
}

// ---------------------------------------------------------------------------
// Kernel 2: build the per-channel LUT. One wave32 per channel.
// 256-bin inclusive scan done on the WMMA unit:
//   cum = H*U + (Lstrict*H)*J     (H = hist as 16x16, flat index 16*M+N)
// Each 16x16x16 product = 4 chained V_WMMA_F32_16X16X4_F32.
// Layouts per CDNA5 ISA 7.12.2:
//   A 16x4 : lane(0..15)=M (lanes 16..31 repeat M), VGPR v holds K = 4c + 2*half + v
//   B 4x16 : lane%16 = N, VGPR v holds row K = 4c + 2*half + v
//   C 16x16: VGPR r holds M = r + 8*half, N = lane%16
// ---------------------------------------------------------------------------
__global__ void che_lut(const int* __restrict__ hist, float* __restrict__ lut) {
    __shared__ int   hi[NBINS];
    __shared__ float hf[NBINS];
    __shared__ float tf[NBINS];
    __shared__ int   sstep[2];      // {step, safe_step}

    const int ch   = blockIdx.x;
    const int lane = threadIdx.x;   // 0..31
    const int half = lane >> 4;     // 0 or 1
    const int nm   = lane & 15;     // N (for B/C) or M (for A)

    #pragma unroll
    for (int k = 0; k < 8; ++k) {
        int i = lane * 8 + k;
        int h = hist[ch * NBINS + i];
        hi[i] = h;
        hf[i] = (float)h;
    }
    __syncthreads();

    v8f acc = {};   // will become row-prefix then full cum
    v8f t   = {};   // Lstrict * H (column prefixes)

    #pragma unroll
    for (int c = 0; c < 4; ++c) {
        const int k0 = 4 * c + 2 * half;
        // acc += H_chunk * U_chunk      (within-row prefix over N)
        v2f aH = { hf[16 * nm + k0], hf[16 * nm + k0 + 1] };
        v2f bU = { (k0     <= nm) ? 1.0f : 0.0f,
                   (k0 + 1 <= nm) ? 1.0f : 0.0f };
        acc = __builtin_amdgcn_wmma_f32_16x16x4_f32(false, aH, false, bU,
                                                    (short)0, acc, false, false);
        // t += Lstrict_chunk * H_chunk  (sum of rows above M, per column)
        v2f aL = { (k0     < nm) ? 1.0f : 0.0f,
                   (k0 + 1 < nm) ? 1.0f : 0.0f };
        v2f bH = { hf[16 * k0 + nm], hf[16 * (k0 + 1) + nm] };
        t = __builtin_amdgcn_wmma_f32_16x16x4_f32(false, aL, false, bH,
                                                  (short)0, t, false, false);
    }

    // Re-layout t (C layout) -> flat LDS so it can be re-fed as an A operand.
    #pragma unroll
    for (int r = 0; r < 8; ++r) {
        int M = r + 8 * half;
        tf[16 * M + nm] = t[r];
    }
    __syncthreads();

    // acc += T * J  (row-sums of T broadcast across N => cross-row offsets)
    const v2f bJ = { 1.0f, 1.0f };
    #pragma unroll
    for (int c = 0; c < 4; ++c) {
        const int k0 = 4 * c + 2 * half;
        v2f aT = { tf[16 * nm + k0], tf[16 * nm + k0 + 1] };
        acc = __builtin_amdgcn_wmma_f32_16x16x4_f32(false, aT, false, bJ,
                                                    (short)0, acc, false, false);
    }

    // Scalar per-channel stats (tiny: 48 channels total).
    if (lane == 0) {
        int total = 0;
        for (int i = 0; i < NBINS; ++i) total += hi[i];
        int li = 255;
        while (li > 0 && hi[li] == 0) --li;
        int st = (total - hi[li]) / 255;
        sstep[0] = st;
        sstep[1] = st > 0 ? st : 1;
    }
    __syncthreads();
    const int st = sstep[0];
    const int ss = sstep[1];

    // lut[i] = clip((excl_cum[i] + ss/2) // ss, 0, 255); step==0 -> identity.
    #pragma unroll
    for (int r = 0; r < 8; ++r) {
        int M    = r + 8 * half;
        int i    = 16 * M + nm;
        int cum  = (int)(acc[r] + 0.5f);        // exact integer in f32
        int excl = cum - hi[i];                 // == cum[i-1] (0 for i==0)
        int l    = (excl + (ss >> 1)) / ss;
        l = imin(l, 255);
        int val  = (st == 0) ? i : l;
        lut[ch * NBINS + i] = (float)val * (1.0f / 255.0f);
    }
}

// ---------------------------------------------------------------------------
// Kernel 3: map pixels through the LUT (LDS-resident) and write output.
// ---------------------------------------------------------------------------
__global__ void che_apply(const float* __restrict__ x,
                          const float* __restrict__ lut,
                          float* __restrict__ out) {
    __shared__ float l[NBINS];
    const int tid = threadIdx.x;
    const int ch  = blockIdx.x / BPC;
    const int blk = blockIdx.x % BPC;

    l[tid] = lut[ch * NBINS + tid];
    __syncthreads();

    const size_t base = (size_t)ch * HW + (size_t)blk * SEG;
    const float4* p = (const float4*)(x + base);
    float4*       q = (float4*)(out + base);

    for (int i = tid; i < NVEC; i += 256) {
        float4 v = p[i];
        float4 o;
        o.x = l[imin(255, (int)(v.x * 255.0f))];
        o.y = l[imin(255, (int)(v.y * 255.0f))];
        o.z = l[imin(255, (int)(v.z * 255.0f))];
        o.w = l[imin(255, (int)(v.w * 255.0f))];
        q[i] = o;
    }
}

// ---------------------------------------------------------------------------
extern "C" void kernel_launch(void* const* d_in, const int* in_sizes, int n_in,
                              void* d_out, int out_size, void* d_ws, size_t ws_size,
                              hipStream_t stream) {
    (void)in_sizes; (void)n_in; (void)out_size; (void)ws_size;
    const float* x = (const float*)d_in[0];
    float* out = (float*)d_out;

    int*   hist = (int*)d_ws;                             // 48*256 ints
    float* lut  = (float*)((char*)d_ws + NCH * NBINS * sizeof(int)); // 48*256 floats

    che_zero_hist<<<NCH, NBINS, 0, stream>>>(hist);
    che_hist<<<NCH * BPC, 256, 0, stream>>>(x, hist);
    che_lut<<<NCH, 32, 0, stream>>>(hist, lut);
    che_apply<<<NCH * BPC, 256, 0, stream>>>(x, lut, out);
}